// spatial_att_80178449482541
// MI455X (gfx1250) — compile-verified
//
#include <hip/hip_runtime.h>
#include <hip/hip_bf16.h>

typedef __attribute__((ext_vector_type(2))) float v2f;
typedef __attribute__((ext_vector_type(4))) float v4f;
typedef __attribute__((ext_vector_type(8))) float v8f;

// ---------------------------------------------------------------------------
// Kernel A: per-row mean+max pooling. One wave32 per row (128 ch = 4 f32/lane).
// Writes x[n] = {mean, max} into workspace; x[N] = {0,0} is the zero pad row.
// feats is streamed (no reuse) -> nontemporal loads keep L2 for xp/nbr.
// ---------------------------------------------------------------------------
__global__ __launch_bounds__(256) void pool_meanmax_kernel(
    const float* __restrict__ feats, float2* __restrict__ xp, int n)
{
    const int lane = threadIdx.x & 31;
    const int row  = blockIdx.x * 8 + (threadIdx.x >> 5);

    if (row < n) {
        const v4f* f4 = (const v4f*)(feats + (long long)row * 128);
        v4f v = __builtin_nontemporal_load(f4 + lane);
        float s = v.x + v.y + v.z + v.w;
        float m = fmaxf(fmaxf(v.x, v.y), fmaxf(v.z, v.w));
        #pragma unroll
        for (int o = 16; o > 0; o >>= 1) {
            s += __shfl_xor(s, o, 32);
            m = fmaxf(m, __shfl_xor(m, o, 32));
        }
        if (lane == 0) xp[row] = make_float2(s * (1.0f / 128.0f), m);
    }
    if (blockIdx.x == 0 && threadIdx.x == 0) xp[n] = make_float2(0.0f, 0.0f);
}

// ---------------------------------------------------------------------------
// Kernel B: fused sparse-conv (via V_WMMA_F32_16X16X4_F32) + sigmoid + scale.
// One wave32 per 16-point tile. Branch-free gather staging:
//   phase 1: all 14 nbr-index loads (clamped addresses + cndmask to pad row)
//   phase 2: all 14 xp float2 gathers
//   phase 3: 14 chained WMMAs (K = 56 = 27 taps * 2 ch, zero padded)
// A (16x4/chunk): lane L supplies a[M=L%16][4i + 2*grp + e]  (grp = L>>4).
// B (4x16): weight vector replicated into every column -> all D columns = y.
// D (16x16 f32): VGPR j holds y[j] on lanes 0-15, y[j+8] on lanes 16-31.
// ---------------------------------------------------------------------------
__global__ __launch_bounds__(256) void conv_sigmoid_scale_kernel(
    const float* __restrict__ feats, const float* __restrict__ weight,
    const int* __restrict__ nbr, const float2* __restrict__ xp,
    float* __restrict__ out, int n, int numTiles)
{
    const int tile = blockIdx.x * 8 + (threadIdx.x >> 5);
    if (tile >= numTiles) return;            // whole-wave uniform exit

    const int lane = threadIdx.x & 31;
    const int grp  = lane >> 4;              // K-half selector
    const int m    = lane & 15;              // matrix row within tile
    const int base = tile * 16;
    const int pt   = base + m;               // this lane's A-matrix point
    const bool valid = (pt < n);
    // Clamped row pointer: always safe to dereference, selected out if invalid.
    const int* nrow = nbr + (long long)(valid ? pt : (n - 1)) * 27;

    // Phase 1: all neighbor indices in flight (no exec divergence, no waits
    // between issues; invalid slots redirected to the zero pad row n).
    int idx[14];
    #pragma unroll
    for (int i = 0; i < 14; ++i) {
        const int k  = 2 * i + grp;          // tap index carried by this lane
        const int kc = k < 27 ? k : 26;      // safe address
        const int id = nrow[kc];
        idx[i] = (valid && k < 27) ? id : n; // cndmask, not branch
    }

    // Phase 2: all pooled-stat gathers in flight (xp is L2-resident, 3.2 MB).
    float2 g[14];
    #pragma unroll
    for (int i = 0; i < 14; ++i) g[i] = xp[idx[i]];

    // Phase 3: chained WMMAs.
    v8f acc = {0.f, 0.f, 0.f, 0.f, 0.f, 0.f, 0.f, 0.f};
    #pragma unroll
    for (int i = 0; i < 14; ++i) {
        const int k  = 2 * i + grp;
        const int kc = k < 27 ? k : 26;
        const float msk = (k < 27) ? 1.0f : 0.0f;
        v2f a, b;
        a.x = g[i].x;
        a.y = g[i].y;
        b.x = weight[2 * kc] * msk;          // B column = w, replicated
        b.y = weight[2 * kc + 1] * msk;
        acc = __builtin_amdgcn_wmma_f32_16x16x4_f32(
            /*neg_a=*/false, a, /*neg_b=*/false, b,
            /*c_mod=*/(short)0, acc, /*reuse_a=*/false, /*reuse_b=*/false);
    }

    // Sigmoid of the 8 y-values this lane holds (M = j + 8*grp).
    float s[8];
    #pragma unroll
    for (int j = 0; j < 8; ++j)
        s[j] = 1.0f / (1.0f + __expf(-acc[j]));

    // Scale pass: half-wave per point, 8 channels (2 x float4) per lane,
    // coalesced 512B rows; streaming traffic marked nontemporal.
    #pragma unroll
    for (int j = 0; j < 8; ++j) {
        const int p = base + j + (grp ? 8 : 0);
        if (p < n) {
            const v4f* f4 = (const v4f*)(feats + (long long)p * 128) + m * 2;
            v4f*       o4 = (v4f*)(out + (long long)p * 128) + m * 2;
            v4f v0 = __builtin_nontemporal_load(f4);
            v4f v1 = __builtin_nontemporal_load(f4 + 1);
            const float sj = s[j];
            v0 *= sj;
            v1 *= sj;
            __builtin_nontemporal_store(v0, o4);
            __builtin_nontemporal_store(v1, o4 + 1);
        }
    }
}

extern "C" void kernel_launch(void* const* d_in, const int* in_sizes, int n_in,
                              void* d_out, int out_size, void* d_ws, size_t ws_size,
                              hipStream_t stream) {
    const float* feats  = (const float*)d_in[0];
    // d_in[1] = coords (unused: kernel map is precomputed in nbr)
    const float* weight = (const float*)d_in[2];
    const int*   nbr    = (const int*)d_in[3];
    float*       out    = (float*)d_out;

    const int C = 128;
    const int n = in_sizes[0] / C;           // 400000

    float2* xp = (float2*)d_ws;              // (n+1) float2, ~3.2 MB

    const int poolBlocks = (n + 7) / 8;      // 8 rows (waves) per 256-thr block
    pool_meanmax_kernel<<<poolBlocks, 256, 0, stream>>>(feats, xp, n);

    const int numTiles = (n + 15) / 16;      // 16-point WMMA tiles
    const int convBlocks = (numTiles + 7) / 8;
    conv_sigmoid_scale_kernel<<<convBlocks, 256, 0, stream>>>(
        feats, weight, nbr, xp, out, n, numTiles);
}